// MaxConv2d_36790689858129
// MI455X (gfx1250) — compile-verified
//
#include <hip/hip_runtime.h>
#include <stdint.h>

// ---------------------------------------------------------------------------
// Max-plus 3x3 "convolution" (tropical semiring). WMMA is inapplicable (no
// max-reduce on matrix cores), so we optimize the VALU path (independent
// v_add_f32 / v_max_num_f32 chains -> VOPD dual-issue) and use the CDNA5
// async global->LDS pipeline (ASYNCcnt) for input staging, with wave-uniform
// scalar (SMEM) weight loads.
// Shapes fixed by the harness: B=8, Cin=Cout=64, H=W=64, k=3, pad=1.
// ---------------------------------------------------------------------------

#define NEGV (-1e30f)

#if __has_builtin(__builtin_amdgcn_global_load_async_to_lds_b128)
#define HAVE_ASYNC_LDS 1
#else
#define HAVE_ASYNC_LDS 0
#endif

typedef int v4i __attribute__((vector_size(16)));
#if HAVE_ASYNC_LDS
typedef __attribute__((address_space(1))) v4i g1_v4i;
typedef __attribute__((address_space(3))) v4i l3_v4i;
#endif

__device__ __forceinline__ void wait_async_zero() {
#if HAVE_ASYNC_LDS
#if __has_builtin(__builtin_amdgcn_s_wait_asynccnt)
  __builtin_amdgcn_s_wait_asynccnt(0);
#else
  asm volatile("s_wait_asynccnt 0" ::: "memory");
#endif
#endif
}

// LDS layout: guard(4 floats, NEG) + 192 rows (r*64+ci) of [64 data | 4 NEG pad],
// row stride 68 floats -> every vec4 slot is 16B aligned, reads are
// conflict-free, and index (base-1) / (base+64) hit NEG pads = -1e30 padding.
#define ROWS   192          // 3 planes * 64 ci
#define RSTR   68
#define XS_LEN (4 + ROWS * RSTR)

template <bool PACKED>
__global__ __launch_bounds__(256) void maxconv_kernel(
    const float* __restrict__ x,    // [8][64][64][64]
    const float* __restrict__ wsrc, // PACKED ? wT[ci][tap][co] : w[co][ci][3][3]
    float* __restrict__ out) {      // [8][64][64][64]
  __shared__ float xs[XS_LEN];

  const int tid = threadIdx.x;
  const int bx  = blockIdx.x;      // 0..511
  const int b   = bx >> 6;
  const int ho  = bx & 63;

  // ---- stage input rows (ho-1, ho, ho+1) for all ci into LDS ------------
  const float4 negv = make_float4(NEGV, NEGV, NEGV, NEGV);
  if (tid == 0) *(float4*)&xs[0] = negv;                       // front guard
  for (int rp = tid; rp < ROWS; rp += 256)                     // row pads
    *(float4*)&xs[4 + rp * RSTR + 64] = negv;

  for (int idx = tid; idx < ROWS * 16; idx += 256) {           // 12 iters
    int row = idx >> 4;            // (r*64 + ci)
    int g   = idx & 15;            // vec4 group within the 64-wide row
    int r   = row >> 6;
    int ci  = row & 63;
    int h   = ho + r - 1;
    int li  = 4 + row * RSTR + g * 4;                          // 16B aligned
    if ((unsigned)h < 64u) {
      const float* src = x + (((size_t)(b * 64 + ci) * 64 + h) * 64 + g * 4);
#if HAVE_ASYNC_LDS
      __builtin_amdgcn_global_load_async_to_lds_b128(
          (g1_v4i*)(v4i*)src, (l3_v4i*)(v4i*)&xs[li], 0, 0);
#else
      *(float4*)&xs[li] = *(const float4*)src;
#endif
    } else {
      *(float4*)&xs[li] = negv;    // out-of-image row -> NEG padding
    }
  }
  wait_async_zero();
  __syncthreads();

  // ---- compute: wave -> 8 co channels, lane -> wo {lane, lane+32} -------
  const int lane    = tid & 31;
  const int co_base = __builtin_amdgcn_readfirstlane((int)(tid >> 5)) * 8;

  float acc0[8], acc1[8];
#pragma unroll
  for (int c = 0; c < 8; ++c) { acc0[c] = NEGV; acc1[c] = NEGV; }

#pragma unroll 1
  for (int ci = 0; ci < 64; ++ci) {
    float sw[9][8];
    if (PACKED) {
      const float* wp = wsrc + (size_t)(ci * 9) * 64 + co_base;  // uniform addr
#pragma unroll
      for (int tap = 0; tap < 9; ++tap)
#pragma unroll
        for (int c = 0; c < 8; ++c)
          sw[tap][c] = wp[tap * 64 + c];                         // -> s_load
    } else {
#pragma unroll
      for (int c = 0; c < 8; ++c)
#pragma unroll
        for (int tap = 0; tap < 9; ++tap)
          sw[tap][c] = wsrc[((size_t)(co_base + c) * 64 + ci) * 9 + tap];
    }

#pragma unroll
    for (int i = 0; i < 3; ++i) {
      const int rb = 4 + (i * 64 + ci) * RSTR;
#pragma unroll
      for (int j = 0; j < 3; ++j) {
        const float xv0 = xs[rb + lane + j - 1];       // wo = lane
        const float xv1 = xs[rb + lane + 31 + j];      // wo = lane + 32
        const int tap = i * 3 + j;
#pragma unroll
        for (int c = 0; c < 8; ++c) {                  // independent add/max
          acc0[c] = fmaxf(acc0[c], xv0 + sw[tap][c]);  // chains -> VOPD pairs
          acc1[c] = fmaxf(acc1[c], xv1 + sw[tap][c]);
        }
      }
    }
  }

  // ---- store ------------------------------------------------------------
#pragma unroll
  for (int c = 0; c < 8; ++c) {
    const size_t o = ((size_t)(b * 64 + co_base + c) * 64 + ho) * 64;
    out[o + lane]      = acc0[c];
    out[o + lane + 32] = acc1[c];
  }
}

// Force the hot instantiation to be emitted first so the disasm snippet
// shows the staging (async-to-LDS) + SMEM-weight prologue.
template __global__ void maxconv_kernel<true>(const float* __restrict__,
                                              const float* __restrict__,
                                              float* __restrict__);
template __global__ void maxconv_kernel<false>(const float* __restrict__,
                                               const float* __restrict__,
                                               float* __restrict__);

// Repack weight[co][ci][3][3] -> wT[ci][tap][co] so that for a fixed (ci,tap)
// the 64 output-channel weights are contiguous (enables merged scalar loads).
__global__ __launch_bounds__(256) void maxconv_repack_w(
    const float* __restrict__ w, float* __restrict__ wT) {
  int idx = blockIdx.x * 256 + threadIdx.x;          // 0 .. 36863
  if (idx >= 64 * 9 * 64) return;
  int co  = idx & 63;
  int t   = idx >> 6;
  int tap = t % 9;
  int ci  = t / 9;
  wT[(ci * 9 + tap) * 64 + co] = w[(co * 64 + ci) * 9 + tap];
}

extern "C" void kernel_launch(void* const* d_in, const int* in_sizes, int n_in,
                              void* d_out, int out_size, void* d_ws, size_t ws_size,
                              hipStream_t stream) {
  (void)in_sizes; (void)n_in; (void)out_size;
  const float* x = (const float*)d_in[0];
  const float* w = (const float*)d_in[1];
  float* out = (float*)d_out;

  const size_t wT_bytes = (size_t)64 * 9 * 64 * sizeof(float);  // 144 KB
  if (ws_size >= wT_bytes) {
    float* wT = (float*)d_ws;
    maxconv_repack_w<<<(64 * 9 * 64 + 255) / 256, 256, 0, stream>>>(w, wT);
    maxconv_kernel<true><<<512, 256, 0, stream>>>(x, wT, out);
  } else {
    maxconv_kernel<false><<<512, 256, 0, stream>>>(x, w, out);
  }
}